// RecurrentGCN_77266461655170
// MI455X (gfx1250) — compile-verified
//
#include <hip/hip_runtime.h>

// Problem constants (from reference setup_inputs)
#define Nn    100000
#define Ee    3200000
#define FIN   256
#define Hh    32

typedef float v2f __attribute__((ext_vector_type(2)));
typedef float v8f __attribute__((ext_vector_type(8)));

// V_WMMA_F32_16X16X4_F32 : D(16x16 f32) = A(16x4 f32) x B(4x16 f32) + C
__device__ __forceinline__ v8f wmma4(v2f a, v2f b, v8f c) {
  return __builtin_amdgcn_wmma_f32_16x16x4_f32(
      /*neg_a=*/false, a, /*neg_b=*/false, b,
      /*c_mod=*/(short)0, c, /*reuse_a=*/false, /*reuse_b=*/false);
}

__device__ __forceinline__ float sigmoidf_(float x) {
  return 1.0f / (1.0f + __expf(-x));
}

// ---------------------------------------------------------------- pack weights
__global__ void pack_weights_kernel(const float* __restrict__ Wz, const float* __restrict__ Wr,
                                    const float* __restrict__ Wh, const float* __restrict__ bz,
                                    const float* __restrict__ br, const float* __restrict__ bh,
                                    float* __restrict__ Wcat, float* __restrict__ bcat) {
  int gid = blockIdx.x * blockDim.x + threadIdx.x;
  if (gid < FIN * 96) {
    int k = gid / 96, c = gid - k * 96;
    float v = (c < 32) ? Wz[k * 32 + c] : ((c < 64) ? Wr[k * 32 + c - 32] : Wh[k * 32 + c - 64]);
    Wcat[gid] = v;
  }
  if (gid < 96) {
    bcat[gid] = (gid < 32) ? bz[gid] : ((gid < 64) ? br[gid - 32] : bh[gid - 64]);
  }
}

// ---------------------------------------------------------------- degree / dis
__global__ void zero_deg_kernel(float* __restrict__ deg) {
  int gid = blockIdx.x * blockDim.x + threadIdx.x;
  if (gid < Nn) deg[gid] = 0.0f;
}

__global__ void deg_accum_kernel(const int* __restrict__ dst, const float* __restrict__ w,
                                 float* __restrict__ deg) {
  int gid = blockIdx.x * blockDim.x + threadIdx.x;
  if (gid < Ee) unsafeAtomicAdd(&deg[dst[gid]], w[gid]);
}

__global__ void finalize_dis_kernel(float* __restrict__ deg_dis) {
  int gid = blockIdx.x * blockDim.x + threadIdx.x;
  if (gid < Nn) deg_dis[gid] = rsqrtf(deg_dis[gid] + 1.0f);
}

// ---------------------------------------------------------------- XW = x @ [Wz|Wr|Wh]
// One wave per 16-row tile; K=256 via 64 f32 WMMAs x 6 column tiles.
__global__ void __launch_bounds__(32) gemm_xw_kernel(const float* __restrict__ x,
                                                     const float* __restrict__ Wcat,
                                                     float* __restrict__ XW) {
  __shared__ float Xs[16 * 260];  // 16x256 tile, stride 260 -> conflict-free frag reads
  const int lane = threadIdx.x;
  const int tile = blockIdx.x * 16;

  // stage A tile (coalesced float4)
  for (int i = lane; i < 1024; i += 32) {
    int r = i >> 6, c4 = (i & 63) << 2;
    const float4 v = *reinterpret_cast<const float4*>(x + (size_t)(tile + r) * FIN + c4);
    *reinterpret_cast<float4*>(&Xs[r * 260 + c4]) = v;
  }
  __syncthreads();

  const int m = lane & 15;             // M row (also N col for B frags)
  const int koff = (lane >> 4) << 1;   // lanes 16-31 hold K+2,K+3

  v8f acc[6] = {};
  const float2* arow = reinterpret_cast<const float2*>(&Xs[m * 260]);

#pragma unroll 4
  for (int k0 = 0; k0 < FIN; k0 += 4) {
    const int kk = k0 + koff;
    float2 af = arow[kk >> 1];
    v2f a; a.x = af.x; a.y = af.y;
    const float* brow = Wcat + (size_t)kk * 96 + m;
#pragma unroll
    for (int t = 0; t < 6; ++t) {
      v2f b; b.x = brow[t * 16]; b.y = brow[96 + t * 16];
      acc[t] = wmma4(a, b, acc[t]);
    }
  }

  const int row0 = (lane >> 4) << 3;
#pragma unroll
  for (int t = 0; t < 6; ++t)
#pragma unroll
    for (int v = 0; v < 8; ++v)
      XW[(size_t)(tile + row0 + v) * 96 + t * 16 + m] = acc[t][v];
}

// ---------------------------------------------------------------- AGG init: self loop + bias
__global__ void init_agg_kernel(const float* __restrict__ dis, const float* __restrict__ XW,
                                const float* __restrict__ bcat, float* __restrict__ AGG) {
  size_t gid = (size_t)blockIdx.x * blockDim.x + threadIdx.x;
  if (gid < (size_t)Nn * 96) {
    size_t i = gid / 96;
    int c = (int)(gid - i * 96);
    float d = dis[i];
    AGG[gid] = d * d * XW[gid] + bcat[c];
  }
}

// ---------------------------------------------------------------- edge scatter (L2 atomics)
// One wave per edge: gather 96 floats of XW[src], atomic-add norm*val into AGG[dst].
__global__ void edge_scatter_kernel(const int* __restrict__ src, const int* __restrict__ dst,
                                    const float* __restrict__ w, const float* __restrict__ dis,
                                    const float* __restrict__ XW, float* __restrict__ AGG) {
  const int lane = threadIdx.x & 31;
  const long long e = (((long long)blockIdx.x * blockDim.x) + threadIdx.x) >> 5;
  if (e >= Ee) return;
  const int s = src[e], d = dst[e];
  const float nrm = dis[s] * w[e] * dis[d];
  const size_t bs = (size_t)s * 96, bd = (size_t)d * 96;
#pragma unroll
  for (int i = 0; i < 3; ++i) {
    const int c = lane + i * 32;
    unsafeAtomicAdd(&AGG[bd + c], nrm * XW[bs + c]);
  }
}

// ---------------------------------------------------------------- fused GRU + readout
// One wave per 16 nodes. Z/R/H~ gate GEMMs (K=64) via f32 WMMA, R through LDS,
// GRU update, h store, y = relu(h) @ head_W + head_b via shfl reduction.
__global__ void __launch_bounds__(32) gru_kernel(const float* __restrict__ AGG,
                                                 const float* __restrict__ Hp,
                                                 const float* __restrict__ LzW, const float* __restrict__ Lzb,
                                                 const float* __restrict__ LrW, const float* __restrict__ Lrb,
                                                 const float* __restrict__ LhW, const float* __restrict__ Lhb,
                                                 const float* __restrict__ headW, const float* __restrict__ headb,
                                                 float* __restrict__ y_out, float* __restrict__ h_out) {
  __shared__ float As[16 * 132];  // cols 0-95: AGG(z|r|h), cols 96-127: Hp; stride 132
  __shared__ float Rs[16 * 36];   // R gate, 16x32, stride 36
  const int lane = threadIdx.x;
  const int tile = blockIdx.x * 16;

  for (int idx = lane; idx < 16 * 128; idx += 32) {
    int r = idx >> 7, c = idx & 127;
    float v = (c < 96) ? AGG[(size_t)(tile + r) * 96 + c] : Hp[(size_t)(tile + r) * 32 + (c - 96)];
    As[r * 132 + c] = v;
  }
  __syncthreads();

  const int m = lane & 15;
  const int n = lane & 15;
  const int koff = (lane >> 4) << 1;

  v8f z0 = {}, z1 = {}, r0 = {}, r1 = {};
#pragma unroll 4
  for (int k0 = 0; k0 < 64; k0 += 4) {
    const int kk = k0 + koff;
    // A-frag columns in As: [Agg_gate | Hp]; 96+(kk-32) == 64+kk
    const int colz = (kk < 32) ? kk : (64 + kk);
    const int colr = (kk < 32) ? (32 + kk) : (64 + kk);
    v2f az; az.x = As[m * 132 + colz]; az.y = As[m * 132 + colz + 1];
    v2f ar; ar.x = As[m * 132 + colr]; ar.y = As[m * 132 + colr + 1];
    const float* lz = LzW + kk * 32 + n;
    const float* lr = LrW + kk * 32 + n;
    v2f b;
    b.x = lz[0];  b.y = lz[32];  z0 = wmma4(az, b, z0);
    b.x = lz[16]; b.y = lz[48];  z1 = wmma4(az, b, z1);
    b.x = lr[0];  b.y = lr[32];  r0 = wmma4(ar, b, r0);
    b.x = lr[16]; b.y = lr[48];  r1 = wmma4(ar, b, r1);
  }

  const float bz0 = Lzb[n], bz1 = Lzb[16 + n];
  const float br0 = Lrb[n], br1 = Lrb[16 + n];
  const int row0 = (lane >> 4) << 3;
#pragma unroll
  for (int v = 0; v < 8; ++v) {
    z0[v] = sigmoidf_(z0[v] + bz0);
    z1[v] = sigmoidf_(z1[v] + bz1);
    r0[v] = sigmoidf_(r0[v] + br0);
    r1[v] = sigmoidf_(r1[v] + br1);
    Rs[(row0 + v) * 36 + n]      = r0[v];
    Rs[(row0 + v) * 36 + 16 + n] = r1[v];
  }
  __syncthreads();

  v8f t0 = {}, t1 = {};
#pragma unroll 4
  for (int k0 = 0; k0 < 64; k0 += 4) {
    const int kk = k0 + koff;
    v2f a;
    a.x = As[m * 132 + 64 + kk];      // Agg_h (kk<32) or Hp (kk>=32), same col formula
    a.y = As[m * 132 + 65 + kk];
    if (kk >= 32) {
      a.x *= Rs[m * 36 + kk - 32];
      a.y *= Rs[m * 36 + kk - 31];
    }
    const float* lh = LhW + kk * 32 + n;
    v2f b;
    b.x = lh[0];  b.y = lh[32];  t0 = wmma4(a, b, t0);
    b.x = lh[16]; b.y = lh[48];  t1 = wmma4(a, b, t1);
  }

  const float bh0 = Lhb[n], bh1 = Lhb[16 + n];
  const float hw0 = headW[n], hw1 = headW[16 + n];
  float p[8];
#pragma unroll
  for (int v = 0; v < 8; ++v) {
    const float ht0 = tanhf(t0[v] + bh0);
    const float ht1 = tanhf(t1[v] + bh1);
    const float hp0 = As[(row0 + v) * 132 + 96 + n];
    const float hp1 = As[(row0 + v) * 132 + 112 + n];
    const float h0 = z0[v] * hp0 + (1.0f - z0[v]) * ht0;
    const float h1 = z1[v] * hp1 + (1.0f - z1[v]) * ht1;
    h_out[(size_t)(tile + row0 + v) * 32 + n]      = h0;
    h_out[(size_t)(tile + row0 + v) * 32 + 16 + n] = h1;
    p[v] = fmaxf(h0, 0.0f) * hw0 + fmaxf(h1, 0.0f) * hw1;
  }
  // reduce across the 16 lanes of each half-wave (xor masks < 16 stay in-half)
#pragma unroll
  for (int s = 1; s < 16; s <<= 1)
#pragma unroll
    for (int v = 0; v < 8; ++v)
      p[v] += __shfl_xor(p[v], s, 32);
  if (n == 0) {
    const float hb = headb[0];
#pragma unroll
    for (int v = 0; v < 8; ++v)
      y_out[tile + row0 + v] = p[v] + hb;
  }
}

// ---------------------------------------------------------------- launch
extern "C" void kernel_launch(void* const* d_in, const int* in_sizes, int n_in,
                              void* d_out, int out_size, void* d_ws, size_t ws_size,
                              hipStream_t stream) {
  const float* x    = (const float*)d_in[0];
  const int*   ei   = (const int*)d_in[1];
  const float* ew   = (const float*)d_in[2];
  const float* Hp   = (const float*)d_in[3];
  const float* Wz   = (const float*)d_in[4];
  const float* bz   = (const float*)d_in[5];
  const float* Wr   = (const float*)d_in[6];
  const float* br   = (const float*)d_in[7];
  const float* Wh   = (const float*)d_in[8];
  const float* bh   = (const float*)d_in[9];
  const float* LzW  = (const float*)d_in[10];
  const float* Lzb  = (const float*)d_in[11];
  const float* LrW  = (const float*)d_in[12];
  const float* Lrb  = (const float*)d_in[13];
  const float* LhW  = (const float*)d_in[14];
  const float* Lhb  = (const float*)d_in[15];
  const float* hW   = (const float*)d_in[16];
  const float* hb   = (const float*)d_in[17];

  const int* src = ei;        // edge_index[0]
  const int* dst = ei + Ee;   // edge_index[1]

  float* out_y = (float*)d_out;          // [N]
  float* out_h = out_y + Nn;             // [N,32]

  // workspace layout (floats): dis | Wcat | bcat | XW | AGG  (~77.3 MB total)
  float* ws   = (float*)d_ws;
  float* dis  = ws;                        // N
  float* Wcat = ws + Nn;                   // 256*96 = 24576
  float* bcat = Wcat + FIN * 96;           // 96
  float* XW   = bcat + 96;                 // N*96
  float* AGG  = XW + (size_t)Nn * 96;      // N*96

  // 1) pack GCN weights/biases into [256,96] / [96]
  pack_weights_kernel<<<(FIN * 96 + 255) / 256, 256, 0, stream>>>(Wz, Wr, Wh, bz, br, bh, Wcat, bcat);
  // 2) degree -> dis = rsqrt(deg+1)
  zero_deg_kernel<<<(Nn + 255) / 256, 256, 0, stream>>>(dis);
  deg_accum_kernel<<<Ee / 256, 256, 0, stream>>>(dst, ew, dis);
  finalize_dis_kernel<<<(Nn + 255) / 256, 256, 0, stream>>>(dis);
  // 3) XW = x @ [Wz|Wr|Wh]  (f32 WMMA)
  gemm_xw_kernel<<<Nn / 16, 32, 0, stream>>>(x, Wcat, XW);
  // 4) AGG = dis^2 * XW + b  (self-loop + bias)
  init_agg_kernel<<<(Nn * 96) / 256, 256, 0, stream>>>(dis, XW, bcat, AGG);
  // 5) AGG += sum_e norm * XW[src]  (one wave per edge, L2 atomics)
  edge_scatter_kernel<<<Ee / 8, 256, 0, stream>>>(src, dst, ew, dis, XW, AGG);
  // 6) fused GRU gates + update + readout (f32 WMMA)
  gru_kernel<<<Nn / 16, 32, 0, stream>>>(AGG, Hp, LzW, Lzb, LrW, Lrb, LhW, Lhb, hW, hb, out_y, out_h);
}